// GraphUNet_31808527794889
// MI455X (gfx1250) — compile-verified
//
#include <hip/hip_runtime.h>

typedef float v2f __attribute__((ext_vector_type(2)));
typedef float v8f __attribute__((ext_vector_type(8)));

#define BN_EPS 1e-5f

// ============================================================================
// GEMM: H[n x 64] = X[n x K] @ W[K x 64], K in {64,128} (compile-time).
// One wave32 computes a 16-row x 64-col tile using v_wmma_f32_16x16x4_f32.
// A layout (16x4 f32): lane l in [0,15]: M=l, VGPR0=K0,VGPR1=K1;
//                      lane l in [16,31]: M=l-16, VGPR0=K2,VGPR1=K3.
// B layout (4x16 f32): mirrored (VGPR j, half h -> K = 2h+j, N = lane%16).
// C/D layout (16x16 f32): VGPR i: lanes0-15 M=i, lanes16-31 M=i+8, N=lane%16.
// Out-of-range rows are CLAMPED on load (their results are never stored), so
// the inner loop runs with EXEC all-ones and no per-load predication.
// ============================================================================
template <int K>
__global__ void gemm_wmma_f32(const float* __restrict__ X,
                              const float* __restrict__ W,
                              float* __restrict__ H, int n) {
  const int wave  = threadIdx.x >> 5;
  const int lane  = threadIdx.x & 31;
  const int mtile = blockIdx.x * 4 + wave;
  const int m0    = mtile << 4;
  if (m0 >= n) return;                  // wave-uniform exit
  const int lr = lane & 15;
  const int lh = lane >> 4;
  int row = m0 + lr;
  if (row >= n) row = n - 1;            // clamp: safe load, result discarded
  const float* __restrict__ xp = X + (long long)row * K;

  v8f acc0 = {0,0,0,0,0,0,0,0};
  v8f acc1 = {0,0,0,0,0,0,0,0};
  v8f acc2 = {0,0,0,0,0,0,0,0};
  v8f acc3 = {0,0,0,0,0,0,0,0};

  #pragma unroll 4
  for (int kk = 0; kk < K; kk += 4) {
    const int ka = kk + (lh << 1);
    v2f a = *(const v2f*)(xp + ka);     // one b64 load per lane

    v2f b0, b1, b2, b3;
    b0.x = W[(ka    ) * 64 +  0 + lr];  b0.y = W[(ka + 1) * 64 +  0 + lr];
    b1.x = W[(ka    ) * 64 + 16 + lr];  b1.y = W[(ka + 1) * 64 + 16 + lr];
    b2.x = W[(ka    ) * 64 + 32 + lr];  b2.y = W[(ka + 1) * 64 + 32 + lr];
    b3.x = W[(ka    ) * 64 + 48 + lr];  b3.y = W[(ka + 1) * 64 + 48 + lr];

    acc0 = __builtin_amdgcn_wmma_f32_16x16x4_f32(false, a, false, b0, (short)0, acc0, false, false);
    acc1 = __builtin_amdgcn_wmma_f32_16x16x4_f32(false, a, false, b1, (short)0, acc1, false, false);
    acc2 = __builtin_amdgcn_wmma_f32_16x16x4_f32(false, a, false, b2, (short)0, acc2, false, false);
    acc3 = __builtin_amdgcn_wmma_f32_16x16x4_f32(false, a, false, b3, (short)0, acc3, false, false);
  }

  if (m0 + 16 <= n) {                   // fast path: full tile, no guards
    #pragma unroll
    for (int i = 0; i < 8; ++i) {
      float* o = H + (long long)(m0 + i + (lh << 3)) * 64 + lr;
      o[ 0] = acc0[i];
      o[16] = acc1[i];
      o[32] = acc2[i];
      o[48] = acc3[i];
    }
  } else {
    #pragma unroll
    for (int i = 0; i < 8; ++i) {
      const int m = m0 + i + (lh << 3);
      if (m < n) {
        float* o = H + (long long)m * 64 + lr;
        o[ 0] = acc0[i];
        o[16] = acc1[i];
        o[32] = acc2[i];
        o[48] = acc3[i];
      }
    }
  }
}

// H[n x 64] = X[n x 3] @ W[3 x 64]
__global__ void gemm_in3(const float* __restrict__ X, const float* __restrict__ W,
                         float* __restrict__ H, int n) {
  long long t = (long long)blockIdx.x * blockDim.x + threadIdx.x;
  if (t >= (long long)n * 64) return;
  int i = (int)(t >> 6), c = (int)(t & 63);
  H[t] = X[i*3+0]*W[0*64+c] + X[i*3+1]*W[1*64+c] + X[i*3+2]*W[2*64+c];
}

// H[n x 3] = X[n x 64] @ W[64 x 3]
__global__ void gemm_out3(const float* __restrict__ X, const float* __restrict__ W,
                          float* __restrict__ H, int n) {
  int i = blockIdx.x * blockDim.x + threadIdx.x;
  if (i >= n) return;
  float a0 = 0.f, a1 = 0.f, a2 = 0.f;
  const float* xp = X + (long long)i * 64;
  for (int c = 0; c < 64; ++c) {
    float v = xp[c];
    a0 += v * W[c*3+0]; a1 += v * W[c*3+1]; a2 += v * W[c*3+2];
  }
  H[(long long)i*3+0] = a0; H[(long long)i*3+1] = a1; H[(long long)i*3+2] = a2;
}

// ============================================================================
// GCN plumbing
// ============================================================================
__global__ void deg_init_k(float* deg, int n) {
  int i = blockIdx.x * blockDim.x + threadIdx.x;
  if (i < n) deg[i] = 1.f;
}
__global__ void deg_accum_k(const int* __restrict__ col, const float* __restrict__ ew,
                            float* deg, long long E) {
  long long e = (long long)blockIdx.x * blockDim.x + threadIdx.x;
  if (e >= E) return;
  __builtin_prefetch(col + e + 2048, 0, 1);     // gfx1250 global_prefetch_b8
  float w = ew ? ew[e] : 1.f;
  if (w != 0.f) atomicAdd(&deg[col[e]], w);
}
__global__ void dinv_k(float* deg, int n) {
  int i = blockIdx.x * blockDim.x + threadIdx.x;
  if (i < n) deg[i] = rsqrtf(deg[i]);
}
// agg = H * dinv^2 + bias (self-loop term), 64 channels, float4-vectorized
__global__ void agg_init64_k(const float* __restrict__ H, const float* __restrict__ dinv,
                             const float* __restrict__ bias, float* __restrict__ agg,
                             int n) {
  long long t = (long long)blockIdx.x * blockDim.x + threadIdx.x;
  if (t >= (long long)n * 16) return;
  int i = (int)(t >> 4), c4 = (int)(t & 15);
  float di = dinv[i];
  float s = di * di;
  float4 h = ((const float4*)H)[t];
  float4 b = ((const float4*)bias)[c4];
  float4 o;
  o.x = h.x * s + b.x; o.y = h.y * s + b.y;
  o.z = h.z * s + b.z; o.w = h.w * s + b.w;
  ((float4*)agg)[t] = o;
}
// generic C-channel version (used for C=3 output layer)
__global__ void agg_init_k(const float* __restrict__ H, const float* __restrict__ dinv,
                           const float* __restrict__ bias, float* __restrict__ agg,
                           int n, int C) {
  long long t = (long long)blockIdx.x * blockDim.x + threadIdx.x;
  if (t >= (long long)n * C) return;
  int i = (int)(t / C), c = (int)(t % C);
  float di = dinv[i];
  agg[t] = H[t] * di * di + bias[c];
}
// 16 threads per edge, 4 channels each (64-channel path)
__global__ void edge_scatter64_k(const int* __restrict__ row, const int* __restrict__ col,
                                 const float* __restrict__ ew, const float* __restrict__ dinv,
                                 const float* __restrict__ H, float* __restrict__ agg,
                                 long long E) {
  long long t = (long long)blockIdx.x * blockDim.x + threadIdx.x;
  if (t >= E * 16) return;
  long long e = t >> 4;
  int g = (int)(t & 15);
  if (g == 0) {                                 // stream-ahead prefetch
    __builtin_prefetch(row + e + 1024, 0, 1);
    __builtin_prefetch(col + e + 1024, 0, 1);
    if (ew) __builtin_prefetch(ew + e + 1024, 0, 1);
  }
  float w = ew ? ew[e] : 1.f;
  if (w == 0.f) return;
  int r = row[e], c = col[e];
  w *= dinv[r] * dinv[c];
  float4 h = *(const float4*)(H + (long long)r * 64 + g * 4);   // one b128 load
  float* ap = agg + (long long)c * 64 + g * 4;
  atomicAdd(ap + 0, h.x * w);
  atomicAdd(ap + 1, h.y * w);
  atomicAdd(ap + 2, h.z * w);
  atomicAdd(ap + 3, h.w * w);
}
// 3-channel scatter (final layer, ew implicit ones)
__global__ void edge_scatter3_k(const int* __restrict__ row, const int* __restrict__ col,
                                const float* __restrict__ dinv, const float* __restrict__ H,
                                float* __restrict__ agg, long long E) {
  long long e = (long long)blockIdx.x * blockDim.x + threadIdx.x;
  if (e >= E) return;
  __builtin_prefetch(row + e + 2048, 0, 1);
  __builtin_prefetch(col + e + 2048, 0, 1);
  int r = row[e], c = col[e];
  float w = dinv[r] * dinv[c];
  const float* hp = H + (long long)r * 3;
  float* ap = agg + (long long)c * 3;
  atomicAdd(ap + 0, hp[0] * w);
  atomicAdd(ap + 1, hp[1] * w);
  atomicAdd(ap + 2, hp[2] * w);
}

// ============================================================================
// BatchNorm (biased var, torch-train semantics) + fused ReLU
// ============================================================================
__global__ void bn_stats_k(const float* __restrict__ X, float* __restrict__ mv, int n) {
  int c = blockIdx.x;            // 64 blocks, one channel each
  int tid = threadIdx.x;
  float s = 0.f, s2 = 0.f;
  for (int i = tid; i < n; i += blockDim.x) {
    float v = X[(long long)i * 64 + c];
    s += v; s2 += v * v;
  }
  __shared__ float sh[256], sh2[256];
  sh[tid] = s; sh2[tid] = s2; __syncthreads();
  for (int off = 128; off > 0; off >>= 1) {
    if (tid < off) { sh[tid] += sh[tid + off]; sh2[tid] += sh2[tid + off]; }
    __syncthreads();
  }
  if (tid == 0) {
    float m = sh[0] / n;
    mv[c]      = m;
    mv[64 + c] = sh2[0] / n - m * m;
  }
}
__global__ void bn_apply_relu_k(float* __restrict__ X, const float* __restrict__ mv,
                                const float* __restrict__ g, const float* __restrict__ b,
                                int n) {
  long long t = (long long)blockIdx.x * blockDim.x + threadIdx.x;
  if (t >= (long long)n * 16) return;
  int c4 = (int)(t & 15);
  float4 m  = ((const float4*)mv)[c4];
  float4 v  = ((const float4*)(mv + 64))[c4];
  float4 ga = ((const float4*)g)[c4];
  float4 be = ((const float4*)b)[c4];
  float4 x = ((float4*)X)[t];
  x.x = fmaxf(ga.x * (x.x - m.x) * rsqrtf(v.x + BN_EPS) + be.x, 0.f);
  x.y = fmaxf(ga.y * (x.y - m.y) * rsqrtf(v.y + BN_EPS) + be.y, 0.f);
  x.z = fmaxf(ga.z * (x.z - m.z) * rsqrtf(v.z + BN_EPS) + be.z, 0.f);
  x.w = fmaxf(ga.w * (x.w - m.w) * rsqrtf(v.w + BN_EPS) + be.w, 0.f);
  ((float4*)X)[t] = x;
}
__global__ void relu4_k(float* X, long long n4) {
  long long t = (long long)blockIdx.x * blockDim.x + threadIdx.x;
  if (t >= n4) return;
  float4 x = ((float4*)X)[t];
  x.x = fmaxf(x.x, 0.f); x.y = fmaxf(x.y, 0.f);
  x.z = fmaxf(x.z, 0.f); x.w = fmaxf(x.w, 0.f);
  ((float4*)X)[t] = x;
}

// ============================================================================
// TopK pooling: scores -> sortable keys -> 4-pass radix select -> compaction
// ============================================================================
__global__ void score_key_k(const float* __restrict__ X, const float* __restrict__ w,
                            unsigned* __restrict__ key, int n) {
  int i = blockIdx.x * blockDim.x + threadIdx.x;
  if (i >= n) return;
  float wn = 0.f, s = 0.f;
  const float4* xp = (const float4*)(X + (long long)i * 64);
  const float4* wp = (const float4*)w;
  for (int c = 0; c < 16; ++c) {
    float4 wv = wp[c], xv = xp[c];
    wn += wv.x*wv.x + wv.y*wv.y + wv.z*wv.z + wv.w*wv.w;
    s  += xv.x*wv.x + xv.y*wv.y + xv.z*wv.z + xv.w*wv.w;
  }
  s = tanhf(s * rsqrtf(wn));
  unsigned u = __float_as_uint(s);
  u = (u & 0x80000000u) ? ~u : (u | 0x80000000u);   // monotone-ascending key
  key[i] = u;
}
__global__ void sel_init_k(unsigned* state, int k) {
  state[0] = 0u;                 // prefix
  state[1] = (unsigned)k;        // elements still needed within current bin
}
__global__ void hist_zero_k(unsigned* hist) { hist[threadIdx.x] = 0u; }
__global__ void hist_k(const unsigned* __restrict__ key, unsigned* hist,
                       const unsigned* __restrict__ state, int n, int shift) {
  int i = blockIdx.x * blockDim.x + threadIdx.x;
  if (i >= n) return;
  unsigned himask = (shift == 24) ? 0u : (0xFFFFFFFFu << (shift + 8));
  unsigned ky = key[i];
  if ((ky & himask) == (state[0] & himask))
    atomicAdd(&hist[(ky >> shift) & 255u], 1u);
}
__global__ void select_bin_k(unsigned* state, const unsigned* hist, int shift) {
  unsigned rem = state[1];
  for (int b = 255; b >= 0; --b) {
    unsigned h = hist[b];
    if (rem > h) { rem -= h; }
    else { state[0] |= ((unsigned)b) << shift; state[1] = rem; return; }
  }
}
// Single-block deterministic compaction: all keys > T, then first `rem` keys == T
__global__ __launch_bounds__(1024)
void compact_k(const unsigned* __restrict__ key, const unsigned* __restrict__ state,
               int* __restrict__ idx, int n, int k) {
  const unsigned T = state[0];
  const unsigned rem = state[1];
  const unsigned ngt = (unsigned)k - rem;
  __shared__ unsigned sgt[1024], seq[1024];
  unsigned cgt = 0, ceq = 0;
  const int tid = threadIdx.x;
  for (int base = 0; base < n; base += 1024) {
    int i = base + tid;
    unsigned g = 0, q = 0;
    if (i < n) { unsigned ky = key[i]; g = (ky > T) ? 1u : 0u; q = (ky == T) ? 1u : 0u; }
    sgt[tid] = g; seq[tid] = q; __syncthreads();
    for (int off = 1; off < 1024; off <<= 1) {
      unsigned ag = 0, aq = 0;
      if (tid >= off) { ag = sgt[tid - off]; aq = seq[tid - off]; }
      __syncthreads();
      sgt[tid] += ag; seq[tid] += aq;
      __syncthreads();
    }
    unsigned tgt = sgt[1023], teq = seq[1023];
    if (i < n) {
      if (g) idx[cgt + sgt[tid] - 1] = i;
      else if (q) { unsigned r = ceq + seq[tid] - 1; if (r < rem) idx[ngt + r] = i; }
    }
    cgt += tgt; ceq += teq;
    __syncthreads();
  }
}
__global__ void zero_i32_k(int* a, int n) {
  int i = blockIdx.x * blockDim.x + threadIdx.x;
  if (i < n) a[i] = 0;
}
__global__ void mark_k(const int* __restrict__ idx, int* __restrict__ newid,
                       int* __restrict__ sel, int k) {
  int j = blockIdx.x * blockDim.x + threadIdx.x;
  if (j >= k) return;
  int node = idx[j];
  newid[node] = j;
  sel[node] = 1;
}
__global__ void relabel_k(const int* __restrict__ row, const int* __restrict__ col,
                          const float* __restrict__ ew, const int* __restrict__ sel,
                          const int* __restrict__ newid, int* __restrict__ nrow,
                          int* __restrict__ ncol, float* __restrict__ new_ew, long long E) {
  long long e = (long long)blockIdx.x * blockDim.x + threadIdx.x;
  if (e >= E) return;
  float w = ew ? ew[e] : 1.f;
  int r = row[e], c = col[e];
  new_ew[e] = (sel[r] && sel[c]) ? w : 0.f;
  nrow[e] = newid[r];
  ncol[e] = newid[c];
}
__global__ void gather_k(const float* __restrict__ X, const int* __restrict__ idx,
                         float* __restrict__ Y, int k) {
  long long t = (long long)blockIdx.x * blockDim.x + threadIdx.x;
  if (t >= (long long)k * 16) return;
  int j = (int)(t >> 4), c4 = (int)(t & 15);
  ((float4*)Y)[t] = ((const float4*)X)[(long long)idx[j] * 16 + c4];
}

// ============================================================================
// Unpool + concat into CAT[pn x 128]
// ============================================================================
__global__ void zero_f32_4k(float* a, long long n4) {
  long long t = (long long)blockIdx.x * blockDim.x + threadIdx.x;
  if (t < n4) ((float4*)a)[t] = make_float4(0.f, 0.f, 0.f, 0.f);
}
__global__ void unpool_k(const float* __restrict__ Xs, const int* __restrict__ idx,
                         float* __restrict__ CAT, int k) {
  long long t = (long long)blockIdx.x * blockDim.x + threadIdx.x;
  if (t >= (long long)k * 16) return;
  int j = (int)(t >> 4), c4 = (int)(t & 15);
  ((float4*)CAT)[(long long)idx[j] * 32 + c4] = ((const float4*)Xs)[t];
}
__global__ void copy_skip_k(const float* __restrict__ xs, float* __restrict__ CAT, int pn) {
  long long t = (long long)blockIdx.x * blockDim.x + threadIdx.x;
  if (t >= (long long)pn * 16) return;
  int i = (int)(t >> 4), c4 = (int)(t & 15);
  ((float4*)CAT)[(long long)i * 32 + 16 + c4] = ((const float4*)xs)[t];
}

// ============================================================================
// Host orchestration
// ============================================================================
static inline size_t align256(size_t x) { return (x + 255) & ~(size_t)255; }
static inline unsigned gblocks(long long work, int bs) {
  return (unsigned)((work + bs - 1) / bs);
}

extern "C" void kernel_launch(void* const* d_in, const int* in_sizes, int n_in,
                              void* d_out, int out_size, void* d_ws, size_t ws_size,
                              hipStream_t stream) {
  (void)n_in; (void)out_size; (void)ws_size;
  const float* x_raw  = (const float*)d_in[0];
  const int*   eidx   = (const int*)  d_in[1];
  const float* W_in   = (const float*)d_in[2];
  const float* b_in   = (const float*)d_in[3];
  const float* dn_W   = (const float*)d_in[4];
  const float* dn_b   = (const float*)d_in[5];
  const float* dn_g   = (const float*)d_in[6];
  const float* dn_be  = (const float*)d_in[7];
  const float* pool_w = (const float*)d_in[8];
  const float* W_bot  = (const float*)d_in[9];
  const float* b_bot  = (const float*)d_in[10];
  const float* up_W   = (const float*)d_in[11];
  const float* up_b   = (const float*)d_in[12];
  const float* up_g   = (const float*)d_in[13];
  const float* up_be  = (const float*)d_in[14];
  const float* W_out  = (const float*)d_in[15];
  const float* b_out  = (const float*)d_in[16];

  const int N = in_sizes[0] / 3;
  const long long E = in_sizes[1] / 2;
  const int ns[4] = { N, N / 2, N / 4, N / 8 };

  // ---- carve workspace ----
  char* p = (char*)d_ws;
  auto take = [&](size_t bytes) -> void* { void* r = p; p += align256(bytes); return r; };
  float*    bufA  = (float*)take((size_t)N * 64 * 4);
  float*    bufB  = (float*)take((size_t)N * 64 * 4);
  float*    Hbuf  = (float*)take((size_t)N * 64 * 4);
  float*    CAT   = (float*)take((size_t)N * 128 * 4);
  float*    dinv  = (float*)take((size_t)N * 4);
  unsigned* key   = (unsigned*)take((size_t)N * 4);
  int*      sel   = (int*)take((size_t)N * 4);
  int*      newid = (int*)take((size_t)N * 4);
  float*    mv    = (float*)take(128 * 4);
  unsigned* state = (unsigned*)take(2 * 4);
  unsigned* hist  = (unsigned*)take(256 * 4);
  int* idxL[3]; float* xsL[3];
  for (int l = 0; l < 3; ++l) idxL[l] = (int*)take((size_t)ns[l + 1] * 4);
  for (int l = 0; l < 3; ++l) xsL[l]  = (float*)take((size_t)ns[l] * 64 * 4);
  const int* rowL[4]; const int* colL[4]; const float* ewL[4];
  int* rowM[4]; int* colM[4]; float* ewM[4];
  rowL[0] = eidx; colL[0] = eidx + E; ewL[0] = nullptr;
  for (int l = 1; l < 4; ++l) {
    rowM[l] = (int*)take((size_t)E * 4);
    colM[l] = (int*)take((size_t)E * 4);
    ewM[l]  = (float*)take((size_t)E * 4);
    rowL[l] = rowM[l]; colL[l] = colM[l]; ewL[l] = ewM[l];
  }

  const int BS = 256;

  // deg/dinv + agg-init + edge scatter for a 64-channel GCN (H already computed)
  auto gcn_tail64 = [&](const float* bias, float* agg, int n, int lvl) {
    deg_init_k<<<gblocks(n, BS), BS, 0, stream>>>(dinv, n);
    deg_accum_k<<<gblocks(E, BS), BS, 0, stream>>>(colL[lvl], ewL[lvl], dinv, E);
    dinv_k<<<gblocks(n, BS), BS, 0, stream>>>(dinv, n);
    agg_init64_k<<<gblocks((long long)n * 16, BS), BS, 0, stream>>>(Hbuf, dinv, bias, agg, n);
    edge_scatter64_k<<<gblocks(E * 16, BS), BS, 0, stream>>>(
        rowL[lvl], colL[lvl], ewL[lvl], dinv, Hbuf, agg, E);
  };
  auto launch_gemm64 = [&](const float* X, const float* W, int n) {
    unsigned mt = (unsigned)((n + 15) / 16);
    gemm_wmma_f32<64><<<(mt + 3) / 4, 128, 0, stream>>>(X, W, Hbuf, n);
  };
  auto launch_gemm128 = [&](const float* X, const float* W, int n) {
    unsigned mt = (unsigned)((n + 15) / 16);
    gemm_wmma_f32<128><<<(mt + 3) / 4, 128, 0, stream>>>(X, W, Hbuf, n);
  };

  float* cur = bufA;
  float* other = bufB;

  // ---- input GCN: x(N x 3) -> cur(N x 64) ----
  gemm_in3<<<gblocks((long long)N * 64, BS), BS, 0, stream>>>(x_raw, W_in, Hbuf, N);
  gcn_tail64(b_in, cur, N, 0);

  // ---- down path ----
  for (int i = 0; i < 3; ++i) {
    const int n = ns[i];
    const int k = ns[i + 1];
    // gcn
    launch_gemm64(cur, dn_W + (size_t)i * 64 * 64, n);
    gcn_tail64(dn_b + i * 64, other, n, i);
    // bn + relu (in place on `other`)
    bn_stats_k<<<64, 256, 0, stream>>>(other, mv, n);
    bn_apply_relu_k<<<gblocks((long long)n * 16, BS), BS, 0, stream>>>(
        other, mv, dn_g + i * 64, dn_be + i * 64, n);
    // stash skip connection
    hipMemcpyAsync(xsL[i], other, (size_t)n * 64 * 4, hipMemcpyDeviceToDevice, stream);
    // topk select
    score_key_k<<<gblocks(n, BS), BS, 0, stream>>>(other, pool_w + i * 64, key, n);
    sel_init_k<<<1, 1, 0, stream>>>(state, k);
    for (int pass = 0; pass < 4; ++pass) {
      int shift = 24 - 8 * pass;
      hist_zero_k<<<1, 256, 0, stream>>>(hist);
      hist_k<<<gblocks(n, BS), BS, 0, stream>>>(key, hist, state, n, shift);
      select_bin_k<<<1, 1, 0, stream>>>(state, hist, shift);
    }
    compact_k<<<1, 1024, 0, stream>>>(key, state, idxL[i], n, k);
    // relabel graph
    zero_i32_k<<<gblocks(n, BS), BS, 0, stream>>>(sel, n);
    zero_i32_k<<<gblocks(n, BS), BS, 0, stream>>>(newid, n);
    mark_k<<<gblocks(k, BS), BS, 0, stream>>>(idxL[i], newid, sel, k);
    relabel_k<<<gblocks(E, BS), BS, 0, stream>>>(
        rowL[i], colL[i], ewL[i], sel, newid, rowM[i + 1], colM[i + 1], ewM[i + 1], E);
    // gather pooled features into the ping-pong partner
    gather_k<<<gblocks((long long)k * 16, BS), BS, 0, stream>>>(other, idxL[i], cur, k);
    // cur now holds level-(i+1) features; `other` free for next round
  }

  // ---- bottom GCN + relu (n3 nodes, level-3 edges) ----
  {
    const int n = ns[3];
    launch_gemm64(cur, W_bot, n);
    gcn_tail64(b_bot, other, n, 3);
    relu4_k<<<gblocks((long long)n * 16, BS), BS, 0, stream>>>(other, (long long)n * 16);
    float* t = cur; cur = other; other = t;
  }

  // ---- up path ----
  for (int i = 0; i < 3; ++i) {
    const int l = 2 - i;          // stack level
    const int pn = ns[l];         // parent node count
    const int k = ns[l + 1];      // current node count
    zero_f32_4k<<<gblocks((long long)pn * 32, BS), BS, 0, stream>>>(CAT, (long long)pn * 32);
    unpool_k<<<gblocks((long long)k * 16, BS), BS, 0, stream>>>(cur, idxL[l], CAT, k);
    copy_skip_k<<<gblocks((long long)pn * 16, BS), BS, 0, stream>>>(xsL[l], CAT, pn);
    // gcn on [pn x 128] with up_W[i] (128 x 64)
    launch_gemm128(CAT, up_W + (size_t)i * 128 * 64, pn);
    gcn_tail64(up_b + i * 64, other, pn, l);
    bn_stats_k<<<64, 256, 0, stream>>>(other, mv, pn);
    bn_apply_relu_k<<<gblocks((long long)pn * 16, BS), BS, 0, stream>>>(
        other, mv, up_g + i * 64, up_be + i * 64, pn);
    float* t = cur; cur = other; other = t;
  }

  // ---- output GCN: cur(N x 64) @ W_out(64 x 3) on level-0 graph ----
  {
    float* out = (float*)d_out;
    gemm_out3<<<gblocks(N, BS), BS, 0, stream>>>(cur, W_out, Hbuf, N);
    deg_init_k<<<gblocks(N, BS), BS, 0, stream>>>(dinv, N);
    deg_accum_k<<<gblocks(E, BS), BS, 0, stream>>>(colL[0], (const float*)nullptr, dinv, E);
    dinv_k<<<gblocks(N, BS), BS, 0, stream>>>(dinv, N);
    agg_init_k<<<gblocks((long long)N * 3, BS), BS, 0, stream>>>(Hbuf, dinv, b_out, out, N, 3);
    edge_scatter3_k<<<gblocks(E, BS), BS, 0, stream>>>(rowL[0], colL[0], dinv, Hbuf, out, E);
  }
}